// MiniRocketFeatureExtractor_3126736192240
// MI455X (gfx1250) — compile-verified
//
#include <hip/hip_runtime.h>

typedef __attribute__((ext_vector_type(2))) float v2f;
typedef __attribute__((ext_vector_type(8))) float v8f;

#define Bn    64
#define Ln    2048
#define Cn    8
#define Kn    84
#define ND    6
#define FPD   20
#define KPAD  96
#define GUARD 128                    // max pad = 4*32
#define XSTR  (Ln + 2*GUARD + 16)    // 2320: guards + bank skew
#define NP    21                     // histogram bins p = 0..20
#define SSTR  32                     // sorted-threshold row stride (padded to 32)
#define BIGF  3.0e38f

__global__ __launch_bounds__(256)
void minirocket_kernel(const float* __restrict__ x,
                       const float* __restrict__ kweights,
                       const float* __restrict__ channel_mask,
                       const float* __restrict__ biases,
                       const float* __restrict__ fmean,
                       const float* __restrict__ fstd,
                       float* __restrict__ out)
{
    __shared__ float xs[Cn][XSTR];            // transposed x[b], zero guards
    __shared__ float bsh[Kn * FPD];           // raw biases for this dilation
    __shared__ float bpad[KPAD * SSTR];       // per-k sorted thresholds, +INF padded
    __shared__ int   ranks[Kn * FPD];         // tie-broken rank of each bias
    __shared__ unsigned int histF[KPAD * NP]; // full-length rank histogram
    __shared__ unsigned int histE[KPAD * NP]; // edge-region rank histogram

    const int tid = threadIdx.x;
    const int b   = blockIdx.x / ND;
    const int di  = blockIdx.x % ND;
    const int d   = 1 << di;                  // dilations 1,2,4,8,16,32
    const int pad = 4 * d;

    // ---- prologue phase 1: init LDS ----
    for (int i = tid; i < Cn * XSTR; i += 256) (&xs[0][0])[i] = 0.f;
    for (int i = tid; i < KPAD * SSTR; i += 256) bpad[i] = BIGF;
    for (int i = tid; i < KPAD * NP; i += 256) { histF[i] = 0u; histE[i] = 0u; }
    for (int i = tid; i < Kn * FPD; i += 256) bsh[i] = biases[di * Kn * FPD + i];
    __syncthreads();

    // ---- prologue phase 2: load x[b] transposed; build ranks + sorted scatter ----
    for (int t = tid; t < Ln; t += 256) {
        const float* xp = x + ((size_t)b * Ln + t) * Cn;
        #pragma unroll
        for (int c = 0; c < Cn; ++c) xs[c][GUARD + t] = xp[c];
    }
    for (int i = tid; i < Kn * FPD; i += 256) {
        const int k = i / FPD, f = i % FPD;
        const float bf = bsh[i];
        int rho = 0;
        #pragma unroll 4
        for (int fq = 0; fq < FPD; ++fq) {
            const float bq = bsh[k * FPD + fq];
            rho += (bq < bf) || ((bq == bf) && (fq < f));
        }
        ranks[i] = rho;
        bpad[k * SSTR + rho] = bf;            // rank is a permutation -> no conflicts
    }
    float wj[9];                              // tap weights (all rows identical)
    #pragma unroll
    for (int j = 0; j < 9; ++j) wj[j] = kweights[j];
    __syncthreads();

    const int lid = tid & 31;
    const int wv  = tid >> 5;
    const int m   = lid & 15;   // M (A) / N (B,D) index within tile
    const int hi  = lid >> 4;   // half-wave select

    // ---- preload A tiles: mask[di][k][c], 6 M-tiles x 2 K-chunks ----
    v2f Areg[6][2];
    #pragma unroll
    for (int mt = 0; mt < 6; ++mt) {
        const int k = mt * 16 + m;
        #pragma unroll
        for (int ch = 0; ch < 2; ++ch) {
            const int c0 = ch * 4 + 2 * hi;
            v2f a;
            if (k < Kn) {
                a.x = channel_mask[(di * Kn + k) * Cn + c0];
                a.y = channel_mask[(di * Kn + k) * Cn + c0 + 1];
            } else { a.x = 0.f; a.y = 0.f; }
            Areg[mt][ch] = a;
        }
    }

    // ---- main loop: this wave owns t in [wv*256, wv*256+256) ----
    const int tw = wv * 256;
    for (int nt = 0; nt < 16; ++nt) {
        const int t0 = tw + nt * 16;
        const int tt = t0 + m;
        const unsigned int eflag = ((tt < pad) || (tt >= Ln - pad)) ? 1u : 0u;

        // B operands: dilated 9-tap weighted sums for this lane's 4 channels
        const int base = GUARD + tt - pad;
        float T0 = 0.f, T1 = 0.f, T2 = 0.f, T3 = 0.f;
        #pragma unroll
        for (int j = 0; j < 9; ++j) {
            const int p = base + j * d;
            T0 = fmaf(wj[j], xs[2 * hi    ][p], T0);
            T1 = fmaf(wj[j], xs[2 * hi + 1][p], T1);
            T2 = fmaf(wj[j], xs[4 + 2 * hi][p], T2);
            T3 = fmaf(wj[j], xs[5 + 2 * hi][p], T3);
        }
        v2f B0; B0.x = T0; B0.y = T1;   // K-chunk 0 (channels 0..3)
        v2f B1; B1.x = T2; B1.y = T3;   // K-chunk 1 (channels 4..7)

        #pragma unroll
        for (int mt = 0; mt < 6; ++mt) {
            v8f acc = {0.f, 0.f, 0.f, 0.f, 0.f, 0.f, 0.f, 0.f};
            acc = __builtin_amdgcn_wmma_f32_16x16x4_f32(
                      false, Areg[mt][0], false, B0, (short)0, acc, false, false);
            acc = __builtin_amdgcn_wmma_f32_16x16x4_f32(
                      false, Areg[mt][1], false, B1, (short)0, acc, false, false);

            // D layout: vgpr r holds M = mt*16 + 8*hi + r, N = m
            #pragma unroll
            for (int r = 0; r < 8; ++r) {
                const int kl = mt * 16 + 8 * hi + r;   // this lane's kernel row
                const float v = acc[r];
                const float* sp = &bpad[kl * SSTR];
                // branchless binary search: p = #{thresholds < v} (rows k>=84 are
                // all +INF -> p==0, lands in ignored histogram rows)
                int p = 0;
                p += (v > sp[p + 15]) ? 16 : 0;
                p += (v > sp[p + 7])  ?  8 : 0;
                p += (v > sp[p + 3])  ?  4 : 0;
                p += (v > sp[p + 1])  ?  2 : 0;
                p += (v > sp[p])      ?  1 : 0;
                atomicAdd(&histF[kl * NP + p], 1u);
                atomicAdd(&histE[kl * NP + p], eflag);  // adds 0 off-edge; no branch
            }
        }
    }
    __syncthreads();

    // ---- finalize: suffix-sum histogram per (k,f), PPV + normalization ----
    const float invL = 1.f / (float)Ln;
    const float invT = 1.f / (float)(Ln - 2 * pad);
    for (int i = tid; i < Kn * FPD; i += 256) {
        const int k   = i / FPD;
        const int rho = ranks[i];
        unsigned int fullc = 0u, edgec = 0u;
        for (int p = rho + 1; p <= 20; ++p) {
            fullc += histF[k * NP + p];
            edgec += histE[k * NP + p];
        }
        const float full = (float)fullc;
        const float edge = (float)edgec;
        const bool  odd  = (((di + k) & 1) == 1);
        const float ppv  = odd ? (full - edge) * invT : full * invL;
        const int g = di * Kn * FPD + i;
        float sd = fstd[g];
        sd = (sd < 1e-6f) ? 1.f : sd;
        out[(size_t)b * (ND * Kn * FPD) + g] = (ppv - fmean[g]) / sd;
    }
}

extern "C" void kernel_launch(void* const* d_in, const int* in_sizes, int n_in,
                              void* d_out, int out_size, void* d_ws, size_t ws_size,
                              hipStream_t stream) {
    const float* x     = (const float*)d_in[0];
    const float* kw    = (const float*)d_in[1];
    const float* cmask = (const float*)d_in[2];
    const float* bias  = (const float*)d_in[3];
    const float* fmean = (const float*)d_in[4];
    const float* fstd  = (const float*)d_in[5];
    float* out = (float*)d_out;
    (void)in_sizes; (void)n_in; (void)d_ws; (void)ws_size; (void)out_size;

    minirocket_kernel<<<dim3(Bn * ND), dim3(256), 0, stream>>>(
        x, kw, cmask, bias, fmean, fstd, out);
}